// DiffAug_25795573580206
// MI455X (gfx1250) — compile-verified
//
#include <hip/hip_runtime.h>
#include <hip/hip_bf16.h>
#include <stdint.h>

// Geometry constants (match reference)
#define B_   8
#define H_   256
#define W_   256
#define D_   12
#define ROW_ (W_ * D_)          // 3072 floats per (b,h) row
#define IMG_ (H_ * ROW_)        // 786432 floats per batch per volume
#define CHUNKS_PER_VB 64
#define ELEMS_PER_CHUNK (IMG_ / CHUNKS_PER_VB)   // 12288

// -------------------------------------------------------------------------
// Pass 1a: partial sums of img0/img1 per (v,b), deterministic tree reduction.
// grid = 16*64 blocks, 256 threads. Each thread: 12 x float4 (48 elems).
// -------------------------------------------------------------------------
__global__ __launch_bounds__(256)
void mean_partial_kernel(const float* __restrict__ img0,
                         const float* __restrict__ img1,
                         double* __restrict__ partials) {
    const int bid   = blockIdx.x;
    const int vb    = bid >> 6;         // 0..15
    const int chunk = bid & 63;         // 0..63
    const int v     = vb >> 3;
    const int b     = vb & 7;
    const float* src = (v == 0 ? img0 : img1)
                     + (size_t)b * IMG_ + (size_t)chunk * ELEMS_PER_CHUNK;
    const int t = threadIdx.x;

    double acc = 0.0;
#pragma unroll
    for (int j = 0; j < 12; ++j) {
        const float4 x = *(const float4*)(src + 4 * (t + 256 * j));
        acc += (double)x.x + (double)x.y + (double)x.z + (double)x.w;
    }

    __shared__ double red[256];
    red[t] = acc;
    __syncthreads();
#pragma unroll
    for (int s = 128; s > 0; s >>= 1) {
        if (t < s) red[t] += red[t + s];
        __syncthreads();
    }
    if (t == 0) partials[bid] = red[0];
}

// -------------------------------------------------------------------------
// Pass 1b: deterministic finalize -> 16 float means in workspace.
// -------------------------------------------------------------------------
__global__ void mean_finalize_kernel(const double* __restrict__ partials,
                                     float* __restrict__ means) {
    const int t = threadIdx.x;
    if (t < 16) {
        double s = 0.0;
        for (int i = 0; i < CHUNKS_PER_VB; ++i) s += partials[t * CHUNKS_PER_VB + i];
        means[t] = (float)(s / (double)IMG_);
    }
}

// -------------------------------------------------------------------------
// Pass 2: one block per output row (v, b, h). grid = 3*B*H = 6144, 256 thr.
// Stage the whole source row (12 KB) into LDS with CDNA5 async global->LDS
// DMA (fire-and-forget, tracked by ASYNCcnt), then emit shifted + affine +
// cutout-masked float4 stores.
// -------------------------------------------------------------------------
__global__ __launch_bounds__(256)
void augment_kernel(const float* __restrict__ img0,
                    const float* __restrict__ img1,
                    const float* __restrict__ seg,
                    const float* __restrict__ bfac,   // [2*B]
                    const float* __restrict__ sfac,   // [2*B]
                    const int*   __restrict__ tx,
                    const int*   __restrict__ ty,
                    const int*   __restrict__ ox,
                    const int*   __restrict__ oy,
                    const float* __restrict__ means,  // [16]
                    float* __restrict__ out) {
    __shared__ float smem[ROW_];

    const int bid = blockIdx.x;          // 0 .. 3*B*H-1
    const int h   = bid & (H_ - 1);
    const int vb  = bid >> 8;            // 0..23
    const int b   = vb & 7;
    const int v   = vb >> 3;             // 0,1,2
    const int t   = threadIdx.x;

    float* orow = out + (size_t)bid * ROW_;

    // translation
    const int txb   = tx[b] - 32;
    const int src_h = h + txb;

    // cutout rectangle (output coordinates)
    const int oxb = ox[b], oyb = oy[b];
    const int lo_x = max(oxb - 64, 0), hi_x = min(oxb + 63, H_ - 1);
    const int lo_y = max(oyb - 64, 0), hi_y = min(oyb + 63, W_ - 1);
    const bool cut_row = (h >= lo_x) && (h <= hi_x);

    // Row fully shifted out of range -> pure zero fill (block-uniform branch)
    if ((unsigned)src_h >= (unsigned)H_) {
        const float4 z = make_float4(0.f, 0.f, 0.f, 0.f);
#pragma unroll
        for (int i = 0; i < 3; ++i)
            *(float4*)(orow + 4 * (t + 256 * i)) = z;
        return;
    }

    const float* srcv = (v == 0) ? img0 : (v == 1) ? img1 : seg;
    const float* srow = srcv + ((size_t)b * H_ + (size_t)src_h) * ROW_;

    // per-(v,b) affine: y = x*s2 + off  (seg: identity, exact)
    float s2 = 1.0f, off = 0.0f;
    if (v < 2) {
        const float m = means[v * 8 + b];
        s2  = 2.0f * sfac[v * 8 + b];
        off = m * (1.0f - s2) + bfac[v * 8 + b] - 0.5f;
    }

    // ---- async DMA: stage full source row into LDS (3 x b128 per lane) ----
#pragma unroll
    for (int i = 0; i < 3; ++i) {
        const int e = 4 * (t + 256 * i);
        const unsigned           lds_addr = (unsigned)(uintptr_t)(smem + e);
        const unsigned long long gaddr    = (unsigned long long)(uintptr_t)(srow + e);
        asm volatile("global_load_async_to_lds_b128 %0, %1, off"
                     :: "v"(lds_addr), "v"(gaddr) : "memory");
    }
#if __has_builtin(__builtin_amdgcn_s_wait_asynccnt)
    __builtin_amdgcn_s_wait_asynccnt(0);
#else
    asm volatile("s_wait_asynccnt 0x0" ::: "memory");
#endif
    __syncthreads();

    // ---- shifted read + affine + cutout, 128-bit stores ----
    const int tyb    = ty[b] - 32;
    const int shiftD = tyb * D_;         // multiple of 12 -> keeps 16B alignment
#pragma unroll
    for (int i = 0; i < 3; ++i) {
        const int e = 4 * (t + 256 * i);  // e % 12 in {0,4,8}: float4 in one w
        const int w = e / D_;
        const int src_w = w + tyb;
        float4 y = make_float4(0.f, 0.f, 0.f, 0.f);
        if ((unsigned)src_w < (unsigned)W_) {
            const float4 x = *(const float4*)(smem + (e + shiftD));
            y.x = fmaf(x.x, s2, off);
            y.y = fmaf(x.y, s2, off);
            y.z = fmaf(x.z, s2, off);
            y.w = fmaf(x.w, s2, off);
        }
        if (cut_row && (w >= lo_y) && (w <= hi_y))
            y = make_float4(0.f, 0.f, 0.f, 0.f);
        *(float4*)(orow + e) = y;
    }
}

// -------------------------------------------------------------------------
extern "C" void kernel_launch(void* const* d_in, const int* in_sizes, int n_in,
                              void* d_out, int out_size, void* d_ws, size_t ws_size,
                              hipStream_t stream) {
    const float* img0 = (const float*)d_in[0];
    const float* img1 = (const float*)d_in[1];
    const float* seg  = (const float*)d_in[2];
    const float* bfac = (const float*)d_in[3];   // [2,B,1,1,1,1] -> 16
    const float* sfac = (const float*)d_in[4];   // [2,B,1,1,1,1] -> 16
    // d_in[5] = cfac: exact no-op for C==1
    const int* tx = (const int*)d_in[6];
    const int* ty = (const int*)d_in[7];
    const int* ox = (const int*)d_in[8];
    const int* oy = (const int*)d_in[9];
    float* out = (float*)d_out;

    double* partials = (double*)d_ws;                       // 1024 doubles
    float*  means    = (float*)((char*)d_ws + 1024 * sizeof(double));

    mean_partial_kernel<<<16 * CHUNKS_PER_VB, 256, 0, stream>>>(img0, img1, partials);
    mean_finalize_kernel<<<1, 32, 0, stream>>>(partials, means);
    augment_kernel<<<3 * B_ * H_, 256, 0, stream>>>(
        img0, img1, seg, bfac, sfac, tx, ty, ox, oy, means, out);
}